// MultiBoxLoss_19207093748047
// MI455X (gfx1250) — compile-verified
//
#include <hip/hip_runtime.h>
#include <hip/hip_bf16.h>
#include <math.h>

#define BB 64
#define PP 8732
#define CC 81
#define OO 16
#define NEGPOS 3
#define THRESH 0.5f

typedef __attribute__((ext_vector_type(2))) float v2f;
typedef __attribute__((ext_vector_type(8))) float v8f;

// ---------------------------------------------------------------------------
// Kernel 1: per-batch matching. One block per batch.
// Computes conf_t[b][p], per-batch smooth-L1 loc loss over positives, num_pos.
// ---------------------------------------------------------------------------
__global__ __launch_bounds__(256) void match_kernel(
    const float* __restrict__ loc_data,   // [B,P,4]
    const float* __restrict__ dbox,       // [P,4] center-size
    const float* __restrict__ truths,     // [B,O,4] point form
    const int* __restrict__ labels,       // [B,O]
    int* __restrict__ conf_t,             // [B,P]
    float* __restrict__ lossl_b,          // [B]
    int* __restrict__ npos_b)             // [B]
{
    __shared__ float s_t[OO * 4];
    __shared__ float s_area[OO];
    __shared__ int   s_lab[OO];
    __shared__ unsigned long long s_bp[OO];   // packed per-truth best prior
    __shared__ float s_bto[PP];               // best truth overlap per prior
    __shared__ unsigned short s_bti[PP];      // best truth idx per prior
    __shared__ float s_redf[256];
    __shared__ int   s_redi[256];

    const int b = blockIdx.x, tid = threadIdx.x;

    if (tid < OO) {
        float x1 = truths[((long long)b * OO + tid) * 4 + 0];
        float y1 = truths[((long long)b * OO + tid) * 4 + 1];
        float x2 = truths[((long long)b * OO + tid) * 4 + 2];
        float y2 = truths[((long long)b * OO + tid) * 4 + 3];
        s_t[tid * 4 + 0] = x1; s_t[tid * 4 + 1] = y1;
        s_t[tid * 4 + 2] = x2; s_t[tid * 4 + 3] = y2;
        s_area[tid] = (x2 - x1) * (y2 - y1);
        s_lab[tid] = labels[b * OO + tid];
        s_bp[tid] = 0ULL;
    }
    __syncthreads();

    unsigned long long lbp[OO];
#pragma unroll
    for (int o = 0; o < OO; ++o) lbp[o] = 0ULL;

    for (int p = tid; p < PP; p += 256) {
        float cx = dbox[p * 4 + 0], cy = dbox[p * 4 + 1];
        float w  = dbox[p * 4 + 2], h  = dbox[p * 4 + 3];
        float bx1 = cx - 0.5f * w, by1 = cy - 0.5f * h;
        float bx2 = cx + 0.5f * w, by2 = cy + 0.5f * h;
        float areab = w * h;
        float bestov = -1.0f; int besto = 0;
#pragma unroll
        for (int o = 0; o < OO; ++o) {
            float ix = fminf(s_t[o * 4 + 2], bx2) - fmaxf(s_t[o * 4 + 0], bx1);
            float iy = fminf(s_t[o * 4 + 3], by2) - fmaxf(s_t[o * 4 + 1], by1);
            ix = fmaxf(ix, 0.0f); iy = fmaxf(iy, 0.0f);
            float inter = ix * iy;
            float iou = inter / (s_area[o] + areab - inter);
            if (iou > bestov) { bestov = iou; besto = o; }   // first-max over o
            unsigned fb = __float_as_uint(iou);              // iou >= 0 -> monotone bits
            unsigned long long pk =
                ((unsigned long long)fb << 32) | (unsigned)(PP - 1 - p); // smaller p wins ties
            if (pk > lbp[o]) lbp[o] = pk;
        }
        s_bto[p] = bestov;
        s_bti[p] = (unsigned short)besto;
    }
#pragma unroll
    for (int o = 0; o < OO; ++o)
        atomicMax(&s_bp[o], lbp[o]);
    __syncthreads();

    if (tid == 0) {
        // force-match, sequential over o (last write wins, matches .at[].set)
        for (int o = 0; o < OO; ++o) {
            unsigned pidx = (unsigned)(PP - 1) - (unsigned)(s_bp[o] & 0xFFFFFFFFu);
            s_bto[pidx] = 2.0f;
            s_bti[pidx] = (unsigned short)o;
        }
    }
    __syncthreads();

    float lossl = 0.0f; int np = 0;
    for (int p = tid; p < PP; p += 256) {
        int o = s_bti[p];
        float ov = s_bto[p];
        int cf = (ov < THRESH) ? 0 : (s_lab[o] + 1);
        conf_t[(long long)b * PP + p] = cf;
        if (cf > 0) {
            ++np;
            float x1 = s_t[o * 4 + 0], y1 = s_t[o * 4 + 1];
            float x2 = s_t[o * 4 + 2], y2 = s_t[o * 4 + 3];
            float pcx = dbox[p * 4 + 0], pcy = dbox[p * 4 + 1];
            float pw  = dbox[p * 4 + 2], ph  = dbox[p * 4 + 3];
            float t0 = ((x1 + x2) * 0.5f - pcx) / (0.1f * pw);
            float t1 = ((y1 + y2) * 0.5f - pcy) / (0.1f * ph);
            float t2 = logf((x2 - x1) / pw) / 0.2f;
            float t3 = logf((y2 - y1) / ph) / 0.2f;
            const float* ld = &loc_data[((long long)b * PP + p) * 4];
            float t[4] = {t0, t1, t2, t3};
#pragma unroll
            for (int j = 0; j < 4; ++j) {
                float d = fabsf(ld[j] - t[j]);
                lossl += (d < 1.0f) ? 0.5f * d * d : d - 0.5f;
            }
        }
    }
    s_redf[tid] = lossl; s_redi[tid] = np;
    __syncthreads();
    for (int st = 128; st > 0; st >>= 1) {
        if (tid < st) { s_redf[tid] += s_redf[tid + st]; s_redi[tid] += s_redi[tid + st]; }
        __syncthreads();
    }
    if (tid == 0) { lossl_b[b] = s_redf[0]; npos_b[b] = s_redi[0]; }
}

// ---------------------------------------------------------------------------
// Kernel 2: per-prior cross entropy (online logsumexp), conf_data streamed
// through LDS with gfx1250 async global->LDS copies.
// ---------------------------------------------------------------------------
#define CE_ROWS 128
__global__ __launch_bounds__(CE_ROWS) void ce_kernel(
    const float* __restrict__ conf,       // [B*P, 81]
    const int* __restrict__ conf_t,       // [B*P]
    float* __restrict__ ce)               // [B*P]
{
    __shared__ float s_conf[CE_ROWS * CC];   // 41472 B, 16B-multiple per block
    const int blk = blockIdx.x;
    const long long base = (long long)blk * CE_ROWS * CC;
    const float* src = conf + base;

#if defined(__gfx1250__)
    {
        unsigned lds_base = (unsigned)(unsigned long long)(const void*)&s_conf[0];
        const int nvec = CE_ROWS * CC / 4;   // b128 transfers
        for (int i = threadIdx.x; i < nvec; i += blockDim.x) {
            unsigned long long g = (unsigned long long)(const void*)(src + i * 4);
            unsigned l = lds_base + (unsigned)(i * 16);
            asm volatile("global_load_async_to_lds_b128 %0, %1, off"
                         :: "v"(l), "v"(g) : "memory");
        }
        asm volatile("s_wait_asynccnt 0" ::: "memory");
    }
    __syncthreads();
#else
    for (int i = threadIdx.x; i < CE_ROWS * CC; i += blockDim.x) s_conf[i] = src[i];
    __syncthreads();
#endif

    const int r = blk * CE_ROWS + threadIdx.x;
    const float* row = &s_conf[threadIdx.x * CC];   // stride 81 dwords: conflict-free
    const int tgt = conf_t[r];
    float m = -INFINITY, s = 0.0f, picked = 0.0f;
#pragma unroll 3
    for (int c = 0; c < CC; ++c) {
        float x = row[c];
        if (c == tgt) picked = x;
        if (x > m) { s *= expf(m - x); m = x; }
        s += expf(x - m);
    }
    ce[r] = m + logf(s) - picked;
}

// ---------------------------------------------------------------------------
// Kernel 3: per-batch hard-negative mining via radix selection of top-k CE
// among negatives (rank value = pos ? 0 : ce, all >= 0 so float bits are
// monotone). loss_c_b = sum(pos ce) + sum(top-k neg ce).
// ---------------------------------------------------------------------------
__global__ __launch_bounds__(256) void select_kernel(
    const int* __restrict__ conf_t,
    const float* __restrict__ ce,
    const int* __restrict__ npos_b,
    float* __restrict__ lossc_b)
{
    __shared__ unsigned s_hist[256];
    __shared__ float s_redf[256];
    __shared__ unsigned s_prefix;
    __shared__ int s_kk;

    const int b = blockIdx.x, tid = threadIdx.x;
    const int* ct = conf_t + (long long)b * PP;
    const float* cep = ce + (long long)b * PP;

    int k = npos_b[b] * NEGPOS;
    if (k > PP) k = PP;
    if (tid == 0) { s_prefix = 0u; s_kk = k; }
    __syncthreads();

    if (k > 0) {
        for (int pass = 3; pass >= 0; --pass) {
            const int shift = pass * 8;
            s_hist[tid] = 0u;
            __syncthreads();
            const unsigned pref = s_prefix;
            for (int p = tid; p < PP; p += 256) {
                float r = (ct[p] > 0) ? 0.0f : fmaxf(cep[p], 0.0f);
                unsigned u = __float_as_uint(r);
                bool match = (pass == 3) || ((u >> (shift + 8)) == (pref >> (shift + 8)));
                if (match) atomicAdd(&s_hist[(u >> shift) & 255u], 1u);
            }
            __syncthreads();
            if (tid == 0) {
                unsigned acc = 0; int kk = s_kk;
                for (int bin = 255; bin >= 0; --bin) {
                    unsigned c = s_hist[bin];
                    if (acc + c >= (unsigned)kk) {
                        s_prefix = pref | ((unsigned)bin << shift);
                        s_kk = kk - (int)acc;
                        break;
                    }
                    acc += c;
                }
            }
            __syncthreads();
        }
    }
    const unsigned T = s_prefix;
    const float Tval = __uint_as_float(T);

    float local = 0.0f;
    for (int p = tid; p < PP; p += 256) {
        if (ct[p] > 0) {
            local += cep[p];
        } else if (k > 0) {
            float r = fmaxf(cep[p], 0.0f);
            if (__float_as_uint(r) > T) local += r;
        }
    }
    s_redf[tid] = local;
    __syncthreads();
    for (int st = 128; st > 0; st >>= 1) {
        if (tid < st) s_redf[tid] += s_redf[tid + st];
        __syncthreads();
    }
    if (tid == 0)
        lossc_b[b] = s_redf[0] + ((k > 0) ? (float)s_kk * Tval : 0.0f);
}

// ---------------------------------------------------------------------------
// Kernel 4: final reduction across 64 batches on the matrix pipe.
// V_WMMA_F32_16X16X4_F32 with all-ones B: D[m][n] = a_m + a_{m+16}; summing
// the 8 accumulator VGPRs + one shfl_xor(16) completes an exact f32 wave sum.
// ---------------------------------------------------------------------------
__device__ inline float wave_sum_wmma(float v) {
    v2f a; a[0] = v;    a[1] = 0.0f;
    v2f o; o[0] = 1.0f; o[1] = 1.0f;
    v8f c = {};
    c = __builtin_amdgcn_wmma_f32_16x16x4_f32(
            /*neg_a=*/false, a, /*neg_b=*/false, o,
            /*c_mod=*/(short)0, c, /*reuse_a=*/false, /*reuse_b=*/false);
    float s = c[0] + c[1] + c[2] + c[3] + c[4] + c[5] + c[6] + c[7];
    s += __shfl_xor(s, 16, 32);
    return s;
}

__global__ __launch_bounds__(32) void finalize_kernel(
    const float* __restrict__ lossl_b,
    const float* __restrict__ lossc_b,
    const int* __restrict__ npos_b,
    float* __restrict__ out)
{
    const int lane = threadIdx.x;   // 32 lanes, EXEC all-ones for WMMA
    float ll = lossl_b[lane] + lossl_b[lane + 32];
    float lc = lossc_b[lane] + lossc_b[lane + 32];
    float np = (float)(npos_b[lane] + npos_b[lane + 32]);
    float Sll = wave_sum_wmma(ll);
    float Slc = wave_sum_wmma(lc);
    float N   = wave_sum_wmma(np);
    if (lane == 0) {
        out[0] = Sll / N;
        out[1] = Slc / N;
    }
}

// ---------------------------------------------------------------------------
extern "C" void kernel_launch(void* const* d_in, const int* in_sizes, int n_in,
                              void* d_out, int out_size, void* d_ws, size_t ws_size,
                              hipStream_t stream) {
    const float* loc_data  = (const float*)d_in[0];   // [B,P,4]
    const float* conf_data = (const float*)d_in[1];   // [B,P,C]
    const float* dbox      = (const float*)d_in[2];   // [P,4]
    const float* truths    = (const float*)d_in[3];   // [B,O,4]
    const int*   labels    = (const int*)d_in[4];     // [B,O]
    float* out = (float*)d_out;

    char* w = (char*)d_ws;
    int*   conf_t  = (int*)w;    w += sizeof(int)   * (size_t)BB * PP;
    float* ce      = (float*)w;  w += sizeof(float) * (size_t)BB * PP;
    float* lossl_b = (float*)w;  w += sizeof(float) * BB;
    float* lossc_b = (float*)w;  w += sizeof(float) * BB;
    int*   npos_b  = (int*)w;    w += sizeof(int)   * BB;

    match_kernel<<<BB, 256, 0, stream>>>(loc_data, dbox, truths, labels,
                                         conf_t, lossl_b, npos_b);

    const int nrows = BB * PP;                 // 558848, divisible by CE_ROWS
    ce_kernel<<<nrows / CE_ROWS, CE_ROWS, 0, stream>>>(conf_data, conf_t, ce);

    select_kernel<<<BB, 256, 0, stream>>>(conf_t, ce, npos_b, lossc_b);

    finalize_kernel<<<1, 32, 0, stream>>>(lossl_b, lossc_b, npos_b, out);
}